// Model_34402688041585
// MI455X (gfx1250) — compile-verified
//
#include <hip/hip_runtime.h>

#define NUM_USERS 100000
#define NUM_ITEMS 50000
#define N_NODES   150000
#define DIM       64
#define NNZ       1000000
#define BATCH     4096
#define SCAN_T    1024
#define SCAN_BLOCKS 147   // ceil(150000/1024)

typedef float v2f __attribute__((ext_vector_type(2)));
typedef float v8f __attribute__((ext_vector_type(8)));

// ---------------------------------------------------------------- init:
// emb = concat(user_emb, item_emb); acc = emb; counts = 0
__global__ void init_kernel(const float* __restrict__ ue, const float* __restrict__ ie,
                            float* __restrict__ emb, float* __restrict__ acc,
                            int* __restrict__ counts) {
    int t = blockIdx.x * blockDim.x + threadIdx.x;     // float4 granularity
    if (t < N_NODES) counts[t] = 0;
    if (t < N_NODES * (DIM / 4)) {
        float4 v = (t < NUM_USERS * (DIM / 4))
                       ? ((const float4*)ue)[t]
                       : ((const float4*)ie)[t - NUM_USERS * (DIM / 4)];
        ((float4*)emb)[t] = v;
        ((float4*)acc)[t] = v;
    }
}

// ---------------------------------------------------------------- histogram of rows
__global__ void hist_kernel(const int* __restrict__ rows, int* __restrict__ counts) {
    int t = blockIdx.x * blockDim.x + threadIdx.x;
    if (t < NNZ) atomicAdd(&counts[rows[t]], 1);
}

// ---------------------------------------------------------------- 3-kernel exclusive scan
__global__ void scan_local_kernel(const int* __restrict__ counts, int* __restrict__ row_ptr,
                                  int* __restrict__ blk_sums) {
    __shared__ int buf[SCAN_T];
    int gid = blockIdx.x * SCAN_T + threadIdx.x;
    int v = (gid < N_NODES) ? counts[gid] : 0;
    buf[threadIdx.x] = v;
    __syncthreads();
    for (int off = 1; off < SCAN_T; off <<= 1) {
        int t = (threadIdx.x >= off) ? buf[threadIdx.x - off] : 0;
        __syncthreads();
        buf[threadIdx.x] += t;
        __syncthreads();
    }
    int incl = buf[threadIdx.x];
    if (gid < N_NODES) row_ptr[gid] = incl - v;          // block-local exclusive
    if (threadIdx.x == SCAN_T - 1) blk_sums[blockIdx.x] = incl;
}

__global__ void scan_blocks_kernel(int* __restrict__ blk_sums) {
    __shared__ int buf[256];
    int v = (threadIdx.x < SCAN_BLOCKS) ? blk_sums[threadIdx.x] : 0;
    buf[threadIdx.x] = v;
    __syncthreads();
    for (int off = 1; off < 256; off <<= 1) {
        int t = (threadIdx.x >= off) ? buf[threadIdx.x - off] : 0;
        __syncthreads();
        buf[threadIdx.x] += t;
        __syncthreads();
    }
    if (threadIdx.x < SCAN_BLOCKS) blk_sums[threadIdx.x] = buf[threadIdx.x] - v;  // exclusive
}

__global__ void scan_add_kernel(int* __restrict__ row_ptr, int* __restrict__ cursor,
                                const int* __restrict__ blk_offs) {
    int gid = blockIdx.x * SCAN_T + threadIdx.x;
    if (gid < N_NODES) {
        int p = row_ptr[gid] + blk_offs[blockIdx.x];
        row_ptr[gid] = p;
        cursor[gid]  = p;
    }
    if (gid == 0) row_ptr[N_NODES] = NNZ;
}

// ---------------------------------------------------------------- scatter COO -> CSR
__global__ void scatter_kernel(const int* __restrict__ rows, const int* __restrict__ cols,
                               const float* __restrict__ vals, int* __restrict__ cursor,
                               int* __restrict__ csr_col, float* __restrict__ csr_val) {
    int t = blockIdx.x * blockDim.x + threadIdx.x;
    if (t < NNZ) {
        int r = rows[t];
        int p = atomicAdd(&cursor[r], 1);
        csr_col[p] = cols[t];
        csr_val[p] = vals[t];
    }
}

// ---------------------------------------------------------------- CSR SpMM, one wave32 per row,
// lane owns 2 dims (float2); fused layer accumulation acc += y. No float atomics.
__global__ void spmm_kernel(const int* __restrict__ row_ptr, const int* __restrict__ csr_col,
                            const float* __restrict__ csr_val, const float* __restrict__ x,
                            float* __restrict__ y, float* __restrict__ acc) {
    int row  = (blockIdx.x * blockDim.x + threadIdx.x) >> 5;
    int lane = threadIdx.x & 31;
    if (row >= N_NODES) return;
    int beg = row_ptr[row];
    int end = row_ptr[row + 1];
    const float2* xv = (const float2*)x;
    float2 s = {0.0f, 0.0f};
    int e = beg;
    for (; e + 1 < end; e += 2) {
        int   c0 = csr_col[e],     c1 = csr_col[e + 1];
        float v0 = csr_val[e],     v1 = csr_val[e + 1];
        float2 t0 = xv[c0 * 32 + lane];
        float2 t1 = xv[c1 * 32 + lane];
        s.x = fmaf(v0, t0.x, s.x); s.y = fmaf(v0, t0.y, s.y);
        s.x = fmaf(v1, t1.x, s.x); s.y = fmaf(v1, t1.y, s.y);
    }
    if (e < end) {
        int   c0 = csr_col[e];
        float v0 = csr_val[e];
        float2 t0 = xv[c0 * 32 + lane];
        s.x = fmaf(v0, t0.x, s.x); s.y = fmaf(v0, t0.y, s.y);
    }
    int o = row * 32 + lane;
    ((float2*)y)[o] = s;
    float2 a = ((float2*)acc)[o];
    a.x += s.x; a.y += s.y;
    ((float2*)acc)[o] = a;
}

// ---------------------------------------------------------------- gather + mean scale (1/4)
__global__ void gather_kernel(const float* __restrict__ acc, const int* __restrict__ uidx,
                              const int* __restrict__ iidx, float* __restrict__ out_user,
                              float* __restrict__ out_item) {
    int t = blockIdx.x * blockDim.x + threadIdx.x;     // BATCH*16 float4 slots
    int b = t >> 4, d = t & 15;
    if (b >= BATCH) return;
    const float4* av = (const float4*)acc;
    float4 u = av[uidx[b] * 16 + d];
    u.x *= 0.25f; u.y *= 0.25f; u.z *= 0.25f; u.w *= 0.25f;
    ((float4*)out_user)[b * 16 + d] = u;
    float4 v = av[(NUM_USERS + iidx[b]) * 16 + d];
    v.x *= 0.25f; v.y *= 0.25f; v.z *= 0.25f; v.w *= 0.25f;
    ((float4*)out_item)[b * 16 + d] = v;
}

// ---------------------------------------------------------------- rating = diag(U * I^T) per
// 16-entry batch tile via V_WMMA_F32_16X16X4_F32 (pure fp32 FMA path). One wave per tile.
__global__ void __launch_bounds__(32)
rating_kernel(const float* __restrict__ u, const float* __restrict__ it,
              float* __restrict__ rating) {
    int tile = blockIdx.x;          // BATCH/16 tiles
    int lane = threadIdx.x;         // 0..31, wave32
    int m    = lane & 15;           // A: row M / B: col N (same batch entry)
    int koff = (lane >> 4) << 1;    // lanes 0-15 hold K0,K1; lanes 16-31 hold K2,K3
    const float* urow = u  + (tile * 16 + m) * DIM;
    const float* irow = it + (tile * 16 + m) * DIM;
    v8f c = {};
#pragma unroll
    for (int k = 0; k < DIM; k += 4) {
        v2f a = { urow[k + koff], urow[k + koff + 1] };
        v2f b = { irow[k + koff], irow[k + koff + 1] };
        c = __builtin_amdgcn_wmma_f32_16x16x4_f32(false, a, false, b, (short)0, c,
                                                  false, false);
    }
    // diag(D): element (k,k) lives in VGPR k, lane k (k=0..7) and VGPR k-8, lane k+16 (k=8..15)
    int j = lane & 7;
    float d = c[j];
    if (lane < 8)        rating[tile * 16 + lane]      = d;
    else if (lane >= 24) rating[tile * 16 + lane - 16] = d;
}

// ---------------------------------------------------------------- launcher
extern "C" void kernel_launch(void* const* d_in, const int* in_sizes, int n_in,
                              void* d_out, int out_size, void* d_ws, size_t ws_size,
                              hipStream_t stream) {
    const int*   user_idx = (const int*)d_in[0];
    const int*   item_idx = (const int*)d_in[1];
    const int*   rows     = (const int*)d_in[2];
    const int*   cols     = (const int*)d_in[3];
    const float* values   = (const float*)d_in[4];
    const float* user_emb = (const float*)d_in[5];
    const float* item_emb = (const float*)d_in[6];

    // workspace layout (all 16B-aligned: N_NODES*DIM*4 = 38,400,000 divisible by 16)
    float* emb_a   = (float*)d_ws;
    float* emb_b   = emb_a + (size_t)N_NODES * DIM;
    float* acc     = emb_b + (size_t)N_NODES * DIM;
    int*   row_ptr = (int*)(acc + (size_t)N_NODES * DIM);   // N_NODES+1
    int*   cursor  = row_ptr + (N_NODES + 4);
    int*   counts  = cursor + N_NODES;
    int*   blk     = counts + N_NODES;                      // 256 entries
    int*   csr_col = blk + 256;
    float* csr_val = (float*)(csr_col + NNZ);

    float* out_rating = (float*)d_out;
    float* out_user   = out_rating + BATCH;
    float* out_item   = out_user + (size_t)BATCH * DIM;

    // 1. init embeddings + zero counts
    {
        int total = N_NODES * (DIM / 4);
        init_kernel<<<(total + 255) / 256, 256, 0, stream>>>(user_emb, item_emb,
                                                             emb_a, acc, counts);
    }
    // 2. histogram
    hist_kernel<<<(NNZ + 255) / 256, 256, 0, stream>>>(rows, counts);
    // 3-5. exclusive scan -> row_ptr, cursor
    scan_local_kernel<<<SCAN_BLOCKS, SCAN_T, 0, stream>>>(counts, row_ptr, blk);
    scan_blocks_kernel<<<1, 256, 0, stream>>>(blk);
    scan_add_kernel<<<SCAN_BLOCKS, SCAN_T, 0, stream>>>(row_ptr, cursor, blk);
    // 6. scatter into CSR
    scatter_kernel<<<(NNZ + 255) / 256, 256, 0, stream>>>(rows, cols, values,
                                                          cursor, csr_col, csr_val);
    // 7-9. three fused SpMM + layer-accumulate passes (ping-pong)
    {
        int blocks = (N_NODES * 32 + 255) / 256;
        spmm_kernel<<<blocks, 256, 0, stream>>>(row_ptr, csr_col, csr_val, emb_a, emb_b, acc);
        spmm_kernel<<<blocks, 256, 0, stream>>>(row_ptr, csr_col, csr_val, emb_b, emb_a, acc);
        spmm_kernel<<<blocks, 256, 0, stream>>>(row_ptr, csr_col, csr_val, emb_a, emb_b, acc);
    }
    // 10. gather batch rows, apply 1/(L+1) scaling
    gather_kernel<<<(BATCH * 16 + 255) / 256, 256, 0, stream>>>(acc, user_idx, item_idx,
                                                                out_user, out_item);
    // 11. ratings via fp32 WMMA diag(U I^T)
    rating_kernel<<<BATCH / 16, 32, 0, stream>>>(out_user, out_item, out_rating);
}